// MemoryBank_71090298683628
// MI455X (gfx1250) — compile-verified
//
#include <hip/hip_runtime.h>
#include <math.h>

#define T_TOKENS   8192
#define D_DIM      4096
#define KS         4
#define N_SLOTS    128
#define TOPK       1024
#define NPAIRS     (TOPK * KS)   // 4096
#define ALPHA      0.1f

typedef __attribute__((ext_vector_type(2))) float v2f;
typedef __attribute__((ext_vector_type(8))) float v8f;

// ---------------------------------------------------------------------------
// Kernel 1: importance score per token. One wave (32 lanes) per token,
// 8 waves per 256-thread block. Streams hidden_states once (128 MB).
// ---------------------------------------------------------------------------
__global__ void __launch_bounds__(256) importance_kernel(
    const float4* __restrict__ hs4, const float* __restrict__ aw,
    const float4* __restrict__ W4, const float* __restrict__ b_imp,
    float* __restrict__ imp) {
  const int lane = threadIdx.x & 31;
  const int wv   = threadIdx.x >> 5;
  const int t    = blockIdx.x * 8 + wv;
  const int D4   = D_DIM / 4;  // 1024 float4 per row
  const float4* row = hs4 + (size_t)t * D4;

  float sumsq = 0.f, dotp = 0.f;
  for (int i = lane; i < D4; i += 32) {
    __builtin_prefetch(&row[i + 256], 0, 1);   // global_prefetch_b8
    float4 h = row[i];
    float4 w = W4[i];
    sumsq += h.x * h.x + h.y * h.y + h.z * h.z + h.w * h.w;
    dotp  += h.x * w.x + h.y * w.y + h.z * w.z + h.w * w.w;
  }
  // wave32 butterfly reduction
  #pragma unroll
  for (int m = 16; m >= 1; m >>= 1) {
    sumsq += __shfl_xor(sumsq, m, 32);
    dotp  += __shfl_xor(dotp,  m, 32);
  }
  if (lane == 0) {
    float mag = sqrtf(sumsq);
    float ent = 0.f;
    #pragma unroll
    for (int j = 0; j < KS; ++j) {
      float a = aw[(size_t)t * KS + j];
      ent -= a * logf(a + 1e-8f);
    }
    float surprise = ent / logf((float)KS);
    float learned  = 1.0f / (1.0f + expf(-(dotp + b_imp[0])));
    imp[t] = mag * (1.0f + surprise) + learned;
  }
}

// ---------------------------------------------------------------------------
// Kernel 2: exact top-1024 by rank counting (total order via index tiebreak).
// importance array staged in LDS; compacted write selTok[rank] = t.
// ---------------------------------------------------------------------------
__global__ void __launch_bounds__(256) select_kernel(
    const float* __restrict__ imp, int* __restrict__ selTok) {
  __shared__ float s[T_TOKENS];   // 32 KB
  for (int i = threadIdx.x; i < T_TOKENS; i += blockDim.x) s[i] = imp[i];
  __syncthreads();
  const int t  = blockIdx.x * blockDim.x + threadIdx.x;
  const float my = s[t];
  int rank = 0;
  for (int j = 0; j < T_TOKENS; ++j) {
    float v = s[j];
    rank += (int)((v > my) || (v == my && j < t));
  }
  if (rank < TOPK) selTok[rank] = t;
}

// ---------------------------------------------------------------------------
// Kernel 3: expand selected tokens into (token,slot) pair arrays + slot counts
// via LDS atomics. Single 1024-thread block (32 waves).
// ---------------------------------------------------------------------------
__global__ void __launch_bounds__(1024) pairs_kernel(
    const int* __restrict__ slotIdx, const int* __restrict__ selTok,
    int* __restrict__ pairTok, int* __restrict__ pairSlot,
    int* __restrict__ slotCnt) {
  __shared__ int cnt[N_SLOTS];
  if (threadIdx.x < N_SLOTS) cnt[threadIdx.x] = 0;
  __syncthreads();
  const int k   = threadIdx.x;
  const int tok = selTok[k];
  #pragma unroll
  for (int j = 0; j < KS; ++j) {
    int sl = slotIdx[(size_t)tok * KS + j];
    pairTok[k * KS + j]  = tok;
    pairSlot[k * KS + j] = sl;
    atomicAdd(&cnt[sl], 1);
  }
  __syncthreads();
  if (threadIdx.x < N_SLOTS) slotCnt[threadIdx.x] = cnt[threadIdx.x];
}

// ---------------------------------------------------------------------------
// Kernel 4: scatter-aggregate as one-hot WMMA matmul + fused EMA write.
// slot_sum = A^T(one-hot) x H via V_WMMA_F32_16X16X4_F32.
// One wave owns all 128 slots (8 tiles of 16) x 16 columns; accumulators
// live entirely in 64 VGPRs. 4 waves/block share the pair arrays in LDS.
// Grid: 64 blocks x 128 threads -> 256 column groups covering D=4096.
// ---------------------------------------------------------------------------
__global__ void __launch_bounds__(128) scatter_wmma_kernel(
    const float* __restrict__ hs, const float* __restrict__ mem,
    const int* __restrict__ pairTok, const int* __restrict__ pairSlot,
    const int* __restrict__ slotCnt, float* __restrict__ out) {
  __shared__ int sTok[NPAIRS];    // 16 KB
  __shared__ int sSlot[NPAIRS];   // 16 KB
  for (int i = threadIdx.x; i < NPAIRS; i += blockDim.x) {
    sTok[i]  = pairTok[i];
    sSlot[i] = pairSlot[i];
  }
  __syncthreads();

  const int lane    = threadIdx.x & 31;
  const int wv      = threadIdx.x >> 5;
  const int colgrp  = blockIdx.x * 4 + wv;    // 0..255
  const int colbase = colgrp * 16;
  const int n       = lane & 15;              // N index / M index in A
  const int half    = lane >> 4;              // 0: K=0,1   1: K=2,3

  v8f acc[8];
  #pragma unroll
  for (int tile = 0; tile < 8; ++tile) acc[tile] = (v8f){};

  for (int p = 0; p < NPAIRS; p += 4) {
    const int kA = p + 2 * half;
    const int kB = kA + 1;
    const int tokA = sTok[kA], tokB = sTok[kB];
    const int slA  = sSlot[kA], slB = sSlot[kB];
    // B (4x16): row K at column n; 64B contiguous per half-wave row segment
    v2f b;
    b.x = hs[(size_t)tokA * D_DIM + colbase + n];
    b.y = hs[(size_t)tokB * D_DIM + colbase + n];
    #pragma unroll
    for (int tile = 0; tile < 8; ++tile) {
      // A (16x4) one-hot: A[m][k] = (pairSlot[k] == tile*16 + m)
      v2f a;
      a.x = (slA == tile * 16 + n) ? 1.0f : 0.0f;
      a.y = (slB == tile * 16 + n) ? 1.0f : 0.0f;
      acc[tile] = __builtin_amdgcn_wmma_f32_16x16x4_f32(
          /*neg_a=*/false, a, /*neg_b=*/false, b,
          /*c_mod=*/(short)0, acc[tile], /*reuse_a=*/false, /*reuse_b=*/false);
    }
  }

  // Finalize: agg = sum / max(cnt,1); EMA write; passthrough for idle slots.
  #pragma unroll
  for (int tile = 0; tile < 8; ++tile) {
    #pragma unroll
    for (int r = 0; r < 8; ++r) {
      const int m    = r + 8 * half;          // C/D layout: VGPR r, lane half
      const int slot = tile * 16 + m;
      const int col  = colbase + n;
      const float cnt = (float)slotCnt[slot];
      const float cur = mem[(size_t)slot * D_DIM + col];
      const float agg = acc[tile][r] / fmaxf(cnt, 1.0f);
      out[(size_t)slot * D_DIM + col] =
          (cnt > 0.f) ? (ALPHA * agg + (1.0f - ALPHA) * cur) : cur;
    }
  }
}

// ---------------------------------------------------------------------------
extern "C" void kernel_launch(void* const* d_in, const int* in_sizes, int n_in,
                              void* d_out, int out_size, void* d_ws, size_t ws_size,
                              hipStream_t stream) {
  const float* hs      = (const float*)d_in[0];   // [8192,4096]
  const float* aw      = (const float*)d_in[1];   // [8192,4]
  const float* mem     = (const float*)d_in[2];   // [1,128,4096]
  const float* W_imp   = (const float*)d_in[3];   // [1,4096]
  const float* b_imp   = (const float*)d_in[4];   // [1]
  const int*   slotIdx = (const int*)d_in[5];     // [8192,4]
  float* out = (float*)d_out;                     // [1,128,4096]

  char* ws = (char*)d_ws;
  float* imp     = (float*)(ws);                                  // 8192 f32
  int*   selTok  = (int*)(ws + 32768);                            // 1024 i32
  int*   pairTok = (int*)(ws + 32768 + 4096);                     // 4096 i32
  int*   pairSlot= (int*)(ws + 32768 + 4096 + 16384);             // 4096 i32
  int*   slotCnt = (int*)(ws + 32768 + 4096 + 16384 + 16384);     // 128 i32

  importance_kernel<<<T_TOKENS / 8, 256, 0, stream>>>(
      (const float4*)hs, aw, (const float4*)W_imp, b_imp, imp);
  select_kernel<<<T_TOKENS / 256, 256, 0, stream>>>(imp, selTok);
  pairs_kernel<<<1, TOPK, 0, stream>>>(slotIdx, selTok, pairTok, pairSlot, slotCnt);
  scatter_wmma_kernel<<<64, 128, 0, stream>>>(hs, mem, pairTok, pairSlot, slotCnt, out);
}